// GNNTorso_74036646248576
// MI455X (gfx1250) — compile-verified
//
#include <hip/hip_runtime.h>
#include <hip/hip_bf16.h>

typedef __attribute__((ext_vector_type(2))) float v2f;
typedef __attribute__((ext_vector_type(8))) float v8f;

#define NN  512
#define CC  32
#define LL  4
#define LN_EPS 1e-5f

// ---------------------------------------------------------------------------
// Pre-kernel: per-relation in-degree -> 1/max(deg,1), written to workspace.
// Graph is batch-invariant, so one small block computes it per launch
// (deterministic; no state carried across calls).
// ---------------------------------------------------------------------------
__global__ __launch_bounds__(256) void rgcn_invdeg_kernel(
    const int* __restrict__ dst, int E, float* __restrict__ invdeg)
{
    __shared__ int deg[3 * NN];
    const int tid = threadIdx.x;
    for (int i = tid; i < 3 * NN; i += 256) deg[i] = 0;
    __syncthreads();
    const int tot = 3 * E;
    for (int i = tid; i < tot; i += 256) {
        int r = i / E;
        atomicAdd(&deg[r * NN + dst[i]], 1);
    }
    __syncthreads();
    for (int i = tid; i < 3 * NN; i += 256) {
        int d = deg[i];
        if (d < 1) d = 1;
        invdeg[i] = 1.0f / (float)d;
    }
}

// ---------------------------------------------------------------------------
// GEMM tile helper: acc[4][2] += (diag(scale) * A[wave-owned 64 rows]) @ W
// using V_WMMA_F32_16X16X4_F32 (full fp32 matrix pipe).
// A layout: lane L -> row M = L%16, K = 4*ks + 2*(L/16) + {0,1}
// B layout: lane L -> col N = L%16, same K pair
// ---------------------------------------------------------------------------
__device__ __forceinline__ void gemm_acc(
    const float* __restrict__ A, const float* __restrict__ W,
    const float* __restrict__ scale, int wid, int mlo, int mhalf,
    v8f acc[4][2])
{
    for (int ks = 0; ks < 8; ++ks) {
        const int kk = ks * 4 + mhalf * 2;
        v2f b0, b1;
        b0.x = W[kk * CC + mlo];
        b0.y = W[(kk + 1) * CC + mlo];
        b1.x = W[kk * CC + 16 + mlo];
        b1.y = W[(kk + 1) * CC + 16 + mlo];
#pragma unroll
        for (int mt = 0; mt < 4; ++mt) {
            const int node = wid * 64 + mt * 16 + mlo;
            const float sc = scale ? scale[node] : 1.0f;
            v2f a;
            a.x = A[node * CC + kk] * sc;
            a.y = A[node * CC + kk + 1] * sc;
            acc[mt][0] = __builtin_amdgcn_wmma_f32_16x16x4_f32(
                false, a, false, b0, (short)0, acc[mt][0], false, false);
            acc[mt][1] = __builtin_amdgcn_wmma_f32_16x16x4_f32(
                false, a, false, b1, (short)0, acc[mt][1], false, false);
        }
    }
}

// ---------------------------------------------------------------------------
// Main kernel: one workgroup (256 thr = 8 wave32) per batch element.
// x (512x32 f32) resident in LDS for all 4 layers.
// ---------------------------------------------------------------------------
__global__ __launch_bounds__(256) void rgcn_main_kernel(
    const float* __restrict__ xx, const float* __restrict__ ss,
    const float* __restrict__ Wf, const float* __restrict__ bf,
    const float* __restrict__ convw, const float* __restrict__ convroot,
    const float* __restrict__ convbias, const float* __restrict__ lng,
    const float* __restrict__ lnb, const int* __restrict__ src,
    const int* __restrict__ dst, const float* __restrict__ invdeg,
    float* __restrict__ out, int E)
{
    __shared__ float xs[NN * CC];   // 64 KB: activations
    __shared__ float hs[NN * CC];   // 64 KB: agg scatter buffer / h spill
    __shared__ float wts[CC * CC];  //  4 KB: staged 32x32 weight
    __shared__ float ivd[3 * NN];   //  6 KB: 1/deg per relation
    __shared__ float wfe[6 * CC];   // W_f
    __shared__ float prm[3 * CC];   // b_f | ln_g | ln_b

    const int bid   = blockIdx.x;
    const int tid   = threadIdx.x;
    const int lane  = tid & 31;
    const int wid   = tid >> 5;
    const int mlo   = lane & 15;
    const int mhalf = lane >> 4;

    // stage graph/feature constants
    for (int i = tid; i < 3 * NN; i += 256) ivd[i] = invdeg[i];
    for (int i = tid; i < 6 * CC; i += 256) wfe[i] = Wf[i];
    if (tid < CC) {
        prm[tid]          = bf[tid];
        prm[CC + tid]     = lng[tid];
        prm[2 * CC + tid] = lnb[tid];
    }
    __syncthreads();

    // x0 = feats @ W_f + b_f   (feats = [i/3, j/3, k/3, t/7, v, m])
    const float mfeat = ss[bid] * 0.125f;  // ss[b,0] / T
    for (int idx = tid; idx < NN * CC; idx += 256) {
        const int n = idx >> 5, c = idx & 31;
        const int t = n >> 6, i = (n >> 4) & 3, j = (n >> 2) & 3, k = n & 3;
        const float v = xx[bid * NN + n];
        float a = prm[c];
        a += ((float)i * (1.0f / 3.0f)) * wfe[0 * CC + c];
        a += ((float)j * (1.0f / 3.0f)) * wfe[1 * CC + c];
        a += ((float)k * (1.0f / 3.0f)) * wfe[2 * CC + c];
        a += ((float)t * (1.0f / 7.0f)) * wfe[3 * CC + c];
        a += v * wfe[4 * CC + c];
        a += mfeat * wfe[5 * CC + c];
        xs[idx] = a;
    }
    __syncthreads();

    for (int l = 0; l < LL; ++l) {
        // h accumulators = conv_bias[l] broadcast (C layout: col = nt*16+mlo)
        v8f acc[4][2];
#pragma unroll
        for (int mt = 0; mt < 4; ++mt)
#pragma unroll
            for (int nt = 0; nt < 2; ++nt) {
                const float bv = convbias[l * CC + nt * 16 + mlo];
                acc[mt][nt] = (v8f){bv, bv, bv, bv, bv, bv, bv, bv};
            }

        // zero agg buffer, stage root weights
        for (int i = tid; i < NN * CC; i += 256) hs[i] = 0.0f;
        for (int i = tid; i < CC * CC; i += 256) wts[i] = convroot[l * CC * CC + i];
        __syncthreads();

        // h += x @ conv_root[l]
        gemm_acc(xs, wts, nullptr, wid, mlo, mhalf, acc);
        __syncthreads();

        for (int r = 0; r < 3; ++r) {
            // scatter: agg[dst] += x[src] via LDS float atomics (lane = channel)
            const int* sR = src + r * E;
            const int* dR = dst + r * E;
            for (int e = wid; e < E; e += 8) {
                const int sn = sR[e], dn = dR[e];
                atomicAdd(&hs[dn * CC + lane], xs[sn * CC + lane]);
            }
            for (int i = tid; i < CC * CC; i += 256)
                wts[i] = convw[(l * 3 + r) * CC * CC + i];
            __syncthreads();

            // h += (agg * invdeg_r) @ conv_w[l][r]
            gemm_acc(hs, wts, &ivd[r * NN], wid, mlo, mhalf, acc);
            __syncthreads();

            if (r < 2) {  // re-zero agg for next relation
                for (int i = tid; i < NN * CC; i += 256) hs[i] = 0.0f;
                __syncthreads();
            }
        }

        // spill h (with ReLU) from WMMA C layout into hs
#pragma unroll
        for (int mt = 0; mt < 4; ++mt)
#pragma unroll
            for (int nt = 0; nt < 2; ++nt)
#pragma unroll
                for (int jj = 0; jj < 8; ++jj) {
                    const int node = wid * 64 + mt * 16 + jj + 8 * mhalf;
                    const int col  = nt * 16 + mlo;
                    const float v  = acc[mt][nt][jj];
                    hs[node * CC + col] = v > 0.0f ? v : 0.0f;
                }
        __syncthreads();

        // LayerNorm over C=32 (lane == channel, wave32 butterfly reductions)
        for (int q = 0; q < 64; ++q) {
            const int node = wid * 64 + q;
            const float v = hs[node * CC + lane];
            float s = v;
            for (int off = 16; off; off >>= 1) s += __shfl_xor(s, off, 32);
            const float mu = s * (1.0f / 32.0f);
            const float d = v - mu;
            float s2 = d * d;
            for (int off = 16; off; off >>= 1) s2 += __shfl_xor(s2, off, 32);
            const float var = s2 * (1.0f / 32.0f);
            xs[node * CC + lane] =
                d * rsqrtf(var + LN_EPS) * prm[CC + lane] + prm[2 * CC + lane];
        }
        __syncthreads();
    }

    // Output: frame t=0 (nodes 0..63), axis-means -> (48, 32)
    for (int idx = tid; idx < 48 * CC; idx += 256) {
        const int row = idx >> 5, c = idx & 31;
        const int g = row >> 4, rc = row & 15;
        const int a0 = rc >> 2, b0 = rc & 3;
        float sum = 0.0f;
#pragma unroll
        for (int q = 0; q < 4; ++q) {
            int node;
            if (g == 0)      node = q * 16 + a0 * 4 + b0;  // mean over i: (j,k)
            else if (g == 1) node = a0 * 16 + q * 4 + b0;  // mean over j: (i,k)
            else             node = a0 * 16 + b0 * 4 + q;  // mean over k: (i,j)
            sum += xs[node * CC + c];
        }
        out[bid * 48 * CC + idx] = sum * 0.25f;
    }
}

// ---------------------------------------------------------------------------
extern "C" void kernel_launch(void* const* d_in, const int* in_sizes, int n_in,
                              void* d_out, int out_size, void* d_ws, size_t ws_size,
                              hipStream_t stream) {
    const float* xx       = (const float*)d_in[0];
    const float* ss       = (const float*)d_in[1];
    const float* Wf       = (const float*)d_in[2];
    const float* bf       = (const float*)d_in[3];
    const float* convw    = (const float*)d_in[4];
    const float* convroot = (const float*)d_in[5];
    const float* convbias = (const float*)d_in[6];
    const float* lng      = (const float*)d_in[7];
    const float* lnb      = (const float*)d_in[8];
    const int*   src      = (const int*)d_in[9];
    const int*   dst      = (const int*)d_in[10];
    float*       out      = (float*)d_out;
    float*       invdeg   = (float*)d_ws;   // 3*512 floats

    const int E = in_sizes[9] / 3;   // edges per relation (6144)
    const int B = in_sizes[1];       // batch (ss has B elements)

    rgcn_invdeg_kernel<<<1, 256, 0, stream>>>(dst, E, invdeg);
    rgcn_main_kernel<<<B, 256, 0, stream>>>(xx, ss, Wf, bf, convw, convroot,
                                            convbias, lng, lnb, src, dst,
                                            invdeg, out, E);
    (void)n_in; (void)out_size; (void)ws_size;
}